// EncoderAttention_27075473834021
// MI455X (gfx1250) — compile-verified
//
#include <hip/hip_runtime.h>
#include <hip/hip_bf16.h>
#include <math.h>

typedef __attribute__((ext_vector_type(2))) float v2f;
typedef __attribute__((ext_vector_type(8))) float v8f;
typedef __attribute__((ext_vector_type(4))) unsigned int u32x4;
typedef __attribute__((ext_vector_type(4))) int i32x4;
typedef __attribute__((ext_vector_type(8))) int i32x8;

#define N_NODES 50000
#define N_EDGES 600000
#define CH      128
#define OUTC    16
#define NREL    8
#define NEG_SLOPE 0.2f

// ---- float <-> orderable uint encoding for atomic max over floats ----
__device__ __forceinline__ unsigned f2ord(float f) {
  unsigned u = __float_as_uint(f);
  return (u & 0x80000000u) ? ~u : (u | 0x80000000u);
}
__device__ __forceinline__ float ord2f(unsigned u) {
  return (u & 0x80000000u) ? __uint_as_float(u & 0x7fffffffu) : __uint_as_float(~u);
}

// ---- Tensor Data Mover: 2D tile Global -> LDS (D# per CDNA5 ISA ch.8) ----
// tensor_w/tensor_h are the remaining extents measured from the tile start
// (TDM returns zeros for OOB reads, which we rely on for the M remainder).
__device__ __forceinline__ void tdm_load_2d(unsigned lds_off, const void* gaddr,
                                            unsigned tensor_w, unsigned tensor_h,
                                            unsigned stride_w, unsigned tile_w,
                                            unsigned tile_h) {
  unsigned long long ga = (unsigned long long)(uintptr_t)gaddr;
  u32x4 g0;
  g0[0] = 0x1u;                                       // count=1, user mode
  g0[1] = lds_off;                                    // lds_addr (bytes)
  g0[2] = (unsigned)(ga & 0xffffffffu);               // global_addr[31:0]
  g0[3] = (unsigned)((ga >> 32) & 0x1ffffffu)         // global_addr[56:32]
          | (2u << 30);                               // type=2 (image)
  i32x8 g1;
  g1[0] = (int)(2u << 16);                            // data_size=2 (4 bytes), mask=0
  g1[1] = (int)((tensor_w & 0xffffu) << 16);          // tensor_dim0[15:0]
  g1[2] = (int)((tensor_w >> 16) | ((tensor_h & 0xffffu) << 16)); // dim0 hi | dim1 lo
  g1[3] = (int)((tensor_h >> 16) | (tile_w << 16));   // dim1 hi | tile_dim0
  g1[4] = (int)tile_h;                                // tile_dim1 (tile_dim2=0)
  g1[5] = (int)stride_w;                              // tensor_dim0_stride[31:0]
  g1[6] = 0;                                          // stride0 hi | stride1 lo
  g1[7] = 0;                                          // stride1 hi
  i32x4 z4 = {0, 0, 0, 0};
  i32x8 z8 = {0, 0, 0, 0, 0, 0, 0, 0};
  __builtin_amdgcn_tensor_load_to_lds(g0, g1, z4, z4, z8, 0);
}

__global__ __launch_bounds__(256) void zero_kernel(float* __restrict__ p, long n) {
  long i = (long)blockIdx.x * blockDim.x + threadIdx.x;
  if (i < n) p[i] = 0.0f;
}

// XW[r] = X @ W[r] via V_WMMA_F32_16X16X4_F32.
// Block = 256 threads (8 waves) -> 128x128 output panel, K staged through LDS
// in 64-wide chunks by the Tensor Data Mover. Wave w owns rows [w*16, w*16+16).
// Fused epilogue computes qdot = row.q, kdot = row.k for the attention logits.
__global__ __launch_bounds__(256) void gemm_rel_wmma(
    const float* __restrict__ X, const float* __restrict__ W,
    const float* __restrict__ qvec, const float* __restrict__ kvec,
    float* __restrict__ XW, float* __restrict__ qdot, float* __restrict__ kdot,
    int N, int C, int H) {
  __shared__ __align__(128) float smem[16384];  // 64KB: X chunk | W chunk
  float* Xs = smem;          // 128 rows x 64 cols
  float* Ws = smem + 8192;   // 64 rows x 128 cols

  const int tid = threadIdx.x;
  const int w = tid >> 5;
  const int lane = tid & 31;
  const int m = lane & 15;
  const int kq = (lane >> 4) << 1;   // K sub-offset: 0 or 2
  const int r = blockIdx.z;
  const int row0 = blockIdx.x * 128;

  v8f acc[8] = {};
  float qv[8], kv[8];
#pragma unroll
  for (int t = 0; t < 8; ++t) {
    qv[t] = qvec[t * 16 + m];
    kv[t] = kvec[t * 16 + m];
  }

  for (int kc = 0; kc < C; kc += 64) {
    if (tid < 32) {  // wave 0 issues the TDM transfers (EXEC ignored by TDM)
      tdm_load_2d(0u, X + (size_t)row0 * C + kc,
                  (unsigned)(C - kc), (unsigned)(N - row0), (unsigned)C,
                  64u, 128u);
      tdm_load_2d(32768u, W + ((size_t)r * C + kc) * H,
                  (unsigned)H, (unsigned)(C - kc), (unsigned)H,
                  128u, 64u);
      __builtin_amdgcn_s_wait_tensorcnt(0);
    }
    __syncthreads();
    // The TDM builtin carries the LDS destination only as an integer inside the
    // descriptor, so clang sees no store to smem and would fold the ds_loads to
    // undef. This escape + memory clobber makes the TDM writes visible.
    asm volatile("" : : "r"(Xs), "r"(Ws) : "memory");
#pragma unroll
    for (int k = 0; k < 64; k += 4) {
      v2f a;
      a.x = Xs[(w * 16 + m) * 64 + k + kq];
      a.y = Xs[(w * 16 + m) * 64 + k + kq + 1];
#pragma unroll
      for (int t = 0; t < 8; ++t) {
        v2f b;
        b.x = Ws[(k + kq) * 128 + t * 16 + m];
        b.y = Ws[(k + kq + 1) * 128 + t * 16 + m];
        acc[t] = __builtin_amdgcn_wmma_f32_16x16x4_f32(
            false, a, false, b, (short)0, acc[t], false, false);
      }
    }
    __syncthreads();
  }

  // Epilogue: store panel + fused attention dot products.
  const int rb = (lane >> 4) << 3;
  const int rowBase = row0 + w * 16 + rb;
#pragma unroll
  for (int i = 0; i < 8; ++i) {
    int row = rowBase + i;
    float sq = 0.0f, sk = 0.0f;
#pragma unroll
    for (int t = 0; t < 8; ++t) {
      sq += acc[t][i] * qv[t];
      sk += acc[t][i] * kv[t];
    }
#pragma unroll
    for (int off = 8; off > 0; off >>= 1) {
      sq += __shfl_xor(sq, off, 16);
      sk += __shfl_xor(sk, off, 16);
    }
    if (row < N) {
      float* orow = XW + ((size_t)r * N + row) * H;
#pragma unroll
      for (int t = 0; t < 8; ++t) orow[t * 16 + m] = acc[t][i];
      if (m == 0) {
        qdot[(size_t)r * N + row] = sq;
        kdot[(size_t)r * N + row] = sk;
      }
    }
  }
}

__global__ __launch_bounds__(256) void edge_logit_kernel(
    const int* __restrict__ src, const int* __restrict__ dst,
    const int* __restrict__ et, const float* __restrict__ qdot,
    const float* __restrict__ kdot, float* __restrict__ ea,
    unsigned* __restrict__ amax, int E, int N) {
  int e = blockIdx.x * blockDim.x + threadIdx.x;
  if (e >= E) return;
  int s = src[e], d = dst[e], r = et[e];
  float a = qdot[(size_t)r * N + d] + kdot[(size_t)r * N + s];
  a = (a > 0.0f) ? a : NEG_SLOPE * a;  // leaky_relu
  ea[e] = a;
  atomicMax(&amax[d], f2ord(a));
}

__global__ __launch_bounds__(256) void edge_exp_kernel(
    const int* __restrict__ dst, const unsigned* __restrict__ amax,
    float* __restrict__ ea, float* __restrict__ denom, int E) {
  int e = blockIdx.x * blockDim.x + threadIdx.x;
  if (e >= E) return;
  int d = dst[e];
  float v = expf(ea[e] - ord2f(amax[d]));
  ea[e] = v;
  atomicAdd(&denom[d], v);
}

// One wave per edge: out[dst] += (ea/denom[dst]) * xw[et, src, :]
__global__ __launch_bounds__(256) void aggregate_kernel(
    const int* __restrict__ src, const int* __restrict__ dst,
    const int* __restrict__ et, const float* __restrict__ ea,
    const float* __restrict__ denom, const float* __restrict__ XW,
    float* __restrict__ out, int E, int N, int H) {
  int e = blockIdx.x * (blockDim.x >> 5) + (threadIdx.x >> 5);
  int lane = threadIdx.x & 31;
  if (e >= E) return;
  int s = src[e], d = dst[e], r = et[e];
  float alpha = ea[e] / denom[d];
  const float4 v = ((const float4*)(XW + ((size_t)r * N + s) * H))[lane];
  float* o = out + (size_t)d * H + lane * 4;
  atomicAdd(o + 0, alpha * v.x);
  atomicAdd(o + 1, alpha * v.y);
  atomicAdd(o + 2, alpha * v.z);
  atomicAdd(o + 3, alpha * v.w);
}

__global__ __launch_bounds__(256) void bias_relu_kernel(
    float* __restrict__ h, const float* __restrict__ b, long n, int H) {
  long i = (long)blockIdx.x * blockDim.x + threadIdx.x;
  if (i >= n) return;
  float v = h[i] + b[i & (H - 1)];
  h[i] = v > 0.0f ? v : 0.0f;
}

// Column sums of h [N,H] -> acc[H]
__global__ __launch_bounds__(128) void colsum_kernel(
    const float* __restrict__ h, float* __restrict__ acc, int N, int H) {
  int c = threadIdx.x;
  float s = 0.0f;
  for (int n = blockIdx.x; n < N; n += gridDim.x) s += h[(size_t)n * H + c];
  atomicAdd(&acc[c], s);
}

// g = (colsum/N) @ lin_w + lin_b ; out = log_softmax(g). One wave; lanes 0..15.
__global__ __launch_bounds__(32) void head_kernel(
    const float* __restrict__ csum, const float* __restrict__ lw,
    const float* __restrict__ lb, float* __restrict__ out, int N, int H, int O) {
  int o = threadIdx.x;
  float g;
  if (o < O) {
    float inv_n = 1.0f / (float)N;
    float s = 0.0f;
    for (int hh = 0; hh < H; ++hh) s += (csum[hh] * inv_n) * lw[hh * O + o];
    g = s + lb[o];
  } else {
    g = -INFINITY;
  }
  float mx = g;
#pragma unroll
  for (int off = 8; off > 0; off >>= 1) mx = fmaxf(mx, __shfl_xor(mx, off, 16));
  float ex = (o < O) ? expf(g - mx) : 0.0f;
  float ssum = ex;
#pragma unroll
  for (int off = 8; off > 0; off >>= 1) ssum += __shfl_xor(ssum, off, 16);
  if (o < O) out[o] = g - mx - logf(ssum);
}

static void run_rgat_layer(const float* xin, const float* W, const float* q,
                           const float* kv, const float* b, float* xw,
                           float* qdot, float* kdot, float* ea, unsigned* amax,
                           float* denom, float* hout, const int* src,
                           const int* dst, const int* et, hipStream_t s) {
  dim3 gG((N_NODES + 127) / 128, 1, NREL);
  gemm_rel_wmma<<<gG, 256, 0, s>>>(xin, W, q, kv, xw, qdot, kdot,
                                   N_NODES, CH, CH);

  zero_kernel<<<(N_NODES + 255) / 256, 256, 0, s>>>((float*)amax, N_NODES);
  zero_kernel<<<(N_NODES + 255) / 256, 256, 0, s>>>(denom, N_NODES);

  edge_logit_kernel<<<(N_EDGES + 255) / 256, 256, 0, s>>>(
      src, dst, et, qdot, kdot, ea, amax, N_EDGES, N_NODES);
  edge_exp_kernel<<<(N_EDGES + 255) / 256, 256, 0, s>>>(dst, amax, ea, denom, N_EDGES);

  long nh = (long)N_NODES * CH;
  zero_kernel<<<(unsigned)((nh + 255) / 256), 256, 0, s>>>(hout, nh);
  aggregate_kernel<<<(N_EDGES + 7) / 8, 256, 0, s>>>(
      src, dst, et, ea, denom, xw, hout, N_EDGES, N_NODES, CH);
  bias_relu_kernel<<<(unsigned)((nh + 255) / 256), 256, 0, s>>>(hout, b, nh, CH);
}

extern "C" void kernel_launch(void* const* d_in, const int* in_sizes, int n_in,
                              void* d_out, int out_size, void* d_ws, size_t ws_size,
                              hipStream_t stream) {
  const float* x  = (const float*)d_in[0];
  const float* W1 = (const float*)d_in[1];
  const float* q1 = (const float*)d_in[2];
  const float* k1 = (const float*)d_in[3];
  const float* b1 = (const float*)d_in[4];
  const float* W2 = (const float*)d_in[5];
  const float* q2 = (const float*)d_in[6];
  const float* k2 = (const float*)d_in[7];
  const float* b2 = (const float*)d_in[8];
  const float* lw = (const float*)d_in[9];
  const float* lb = (const float*)d_in[10];
  const int* ei   = (const int*)d_in[11];  // [2, E]
  const int* et   = (const int*)d_in[12];  // [E]
  const int* src = ei;
  const int* dst = ei + N_EDGES;

  // Workspace carve-up (floats)
  float* ws   = (float*)d_ws;
  float* xw   = ws;                                // NREL*N*H
  float* hbuf = xw + (size_t)NREL * N_NODES * CH;  // N*H
  float* qdot = hbuf + (size_t)N_NODES * CH;       // NREL*N
  float* kdot = qdot + (size_t)NREL * N_NODES;     // NREL*N
  float* ea   = kdot + (size_t)NREL * N_NODES;     // E
  unsigned* amax = (unsigned*)(ea + N_EDGES);      // N
  float* denom   = (float*)(amax + N_NODES);       // N
  float* csum    = denom + N_NODES;                // H

  // Layer 1: x -> hbuf
  run_rgat_layer(x, W1, q1, k1, b1, xw, qdot, kdot, ea, amax, denom, hbuf,
                 src, dst, et, stream);
  // Layer 2: hbuf -> hbuf (xw fully recomputed from hbuf before hbuf is zeroed)
  run_rgat_layer(hbuf, W2, q2, k2, b2, xw, qdot, kdot, ea, amax, denom, hbuf,
                 src, dst, et, stream);

  // Head: mean over nodes commutes with the linear layer
  zero_kernel<<<1, 256, 0, stream>>>(csum, CH);
  colsum_kernel<<<256, 128, 0, stream>>>(hbuf, csum, N_NODES, CH);
  head_kernel<<<1, 32, 0, stream>>>(csum, lw, lb, (float*)d_out, N_NODES, CH, OUTC);
}